// GCN_AML_32246614458736
// MI455X (gfx1250) — compile-verified
//
#include <hip/hip_runtime.h>
#include <math.h>

#define N_NODES 100000
#define N_EDGES 1600000
#define HDIM    128
#define CLS     2
#define LAYERS  3
#define BN_EPS  1e-5f

#define N_TILES      (N_NODES / 16)          // 6250 16-row tiles
#define MT_MAIN      4
#define FULL_BLOCKS  (N_TILES / MT_MAIN)     // 1562 blocks of 4 tiles
#define REM_TILES    (N_TILES - FULL_BLOCKS * MT_MAIN)   // 2

typedef __attribute__((ext_vector_type(2))) float v2f;
typedef __attribute__((ext_vector_type(8))) float v8f;

// ---------- degree / normalization ----------
__global__ void k_init_dis(float* dis) {
    int i = blockIdx.x * blockDim.x + threadIdx.x;
    if (i < N_NODES) dis[i] = 1.0f;   // self-loop contributes 1 to every degree
}

__global__ void k_deg(const int* __restrict__ dst, float* __restrict__ dis) {
    int i = blockIdx.x * blockDim.x + threadIdx.x;
    if (i < N_EDGES) unsafeAtomicAdd(&dis[dst[i]], 1.0f);
}

__global__ void k_finish_dis(float* dis) {
    int i = blockIdx.x * blockDim.x + threadIdx.x;
    if (i < N_NODES) dis[i] = rsqrtf(dis[i]);   // deg >= 1 always (self-loop)
}

// ---------- GEMM: hw = h @ W (128x128) via V_WMMA_F32_16X16X4_F32 ----------
// 256 threads = 8 waves; wave w owns output cols [16w,16w+16).
// Each wave carries MT independent 16x16 accumulators (MT row-tiles per block):
//  - B fragment loaded once per K-step, reused MT times
//  - MT WMMAs per step write distinct D regs -> no WMMA->WMMA RAW stall
template <int MT>
__global__ void __launch_bounds__(256) k_gemm128(const float* __restrict__ h,
                                                 const float* __restrict__ W,
                                                 float* __restrict__ hw,
                                                 int tileBase) {
    const int wave    = threadIdx.x >> 5;
    const int lane    = threadIdx.x & 31;
    const int colBase = wave << 4;
    const int mn      = lane & 15;          // A-row / B-col / D-col index for this lane
    const int kOff    = (lane >> 4) << 1;   // 0 (lanes 0-15) or 2 (lanes 16-31)
    const int rowBase = (tileBase + blockIdx.x * MT) << 4;

    const float* arow[MT];
    #pragma unroll
    for (int t = 0; t < MT; ++t)
        arow[t] = h + (size_t)(rowBase + t * 16 + mn) * HDIM;   // A: row per lane
    const float* bcol = W + colBase + mn;                        // B: col per lane

    const v8f zero = {};
    v8f acc[MT];
    #pragma unroll
    for (int t = 0; t < MT; ++t) acc[t] = zero;

    for (int k = 0; k < HDIM; k += 4) {
        v2f b;
        b.x = bcol[(size_t)(k + kOff)     * HDIM];
        b.y = bcol[(size_t)(k + kOff + 1) * HDIM];
        #pragma unroll
        for (int t = 0; t < MT; ++t) {
            v2f a = *(const v2f*)(arow[t] + k + kOff);           // 8B-aligned pair
            acc[t] = __builtin_amdgcn_wmma_f32_16x16x4_f32(
                false, a, false, b, (short)0, acc[t], false, false);
        }
    }

    // D layout: lanes 0-15 -> M = r, lanes 16-31 -> M = r+8; col = lane&15
    const int mBase = (lane >> 4) << 3;
    #pragma unroll
    for (int t = 0; t < MT; ++t) {
        float* drow = hw + (size_t)(rowBase + t * 16 + mBase) * HDIM + colBase + mn;
        #pragma unroll
        for (int r = 0; r < 8; ++r) drow[(size_t)r * HDIM] = acc[t][r];
    }
}

// ---------- scatter init: self-loop contribution + bias ----------
__global__ void k_init_next(const float* __restrict__ hw, const float* __restrict__ dis,
                            const float* __restrict__ bias, float* __restrict__ hnext) {
    size_t idx = (size_t)blockIdx.x * blockDim.x + threadIdx.x;
    if (idx < (size_t)N_NODES * HDIM) {
        int i = (int)(idx >> 7);
        int j = (int)(idx & 127);
        float s = dis[i];                    // self-loop norm = dis[i]^2
        hnext[idx] = bias[j] + hw[idx] * s * s;
    }
}

// ---------- edge scatter: one wave per edge, float4 per lane ----------
__global__ void __launch_bounds__(256) k_scatter(const int* __restrict__ src,
                                                 const int* __restrict__ dst,
                                                 const float* __restrict__ dis,
                                                 const float* __restrict__ hw,
                                                 float* __restrict__ hnext) {
    int gid  = blockIdx.x * blockDim.x + threadIdx.x;
    int e    = gid >> 5;
    int lane = threadIdx.x & 31;
    if (e >= N_EDGES) return;
    int s = src[e], d = dst[e];
    float nrm = dis[s] * dis[d];
    float4 v = ((const float4*)(hw + (size_t)s * HDIM))[lane];
    float* pd = hnext + (size_t)d * HDIM + (lane << 2);
    unsafeAtomicAdd(pd + 0, v.x * nrm);
    unsafeAtomicAdd(pd + 1, v.y * nrm);
    unsafeAtomicAdd(pd + 2, v.z * nrm);
    unsafeAtomicAdd(pd + 3, v.w * nrm);
}

// ---------- BatchNorm (eval) + ReLU, in place ----------
__global__ void k_bn_relu(float* __restrict__ h, const float* __restrict__ mean,
                          const float* __restrict__ var, const float* __restrict__ gamma,
                          const float* __restrict__ beta) {
    size_t idx = (size_t)blockIdx.x * blockDim.x + threadIdx.x;
    if (idx < (size_t)N_NODES * HDIM) {
        int j = (int)(idx & 127);
        float v = (h[idx] - mean[j]) * rsqrtf(var[j] + BN_EPS) * gamma[j] + beta[j];
        h[idx] = fmaxf(v, 0.0f);
    }
}

// ---------- classifier (128 -> 2) + log_softmax, one wave per node ----------
__global__ void __launch_bounds__(256) k_classify(const float* __restrict__ h,
                                                  const float* __restrict__ Wc,
                                                  const float* __restrict__ bc,
                                                  float* __restrict__ out) {
    int gid  = blockIdx.x * blockDim.x + threadIdx.x;
    int node = gid >> 5;
    int lane = threadIdx.x & 31;
    if (node >= N_NODES) return;
    float4 v = ((const float4*)(h + (size_t)node * HDIM))[lane];
    int k0 = lane << 2;
    float d0 = v.x * Wc[(k0    ) * CLS]     + v.y * Wc[(k0 + 1) * CLS]
             + v.z * Wc[(k0 + 2) * CLS]     + v.w * Wc[(k0 + 3) * CLS];
    float d1 = v.x * Wc[(k0    ) * CLS + 1] + v.y * Wc[(k0 + 1) * CLS + 1]
             + v.z * Wc[(k0 + 2) * CLS + 1] + v.w * Wc[(k0 + 3) * CLS + 1];
    #pragma unroll
    for (int off = 16; off > 0; off >>= 1) {
        d0 += __shfl_xor(d0, off, 32);
        d1 += __shfl_xor(d1, off, 32);
    }
    if (lane == 0) {
        float l0 = d0 + bc[0], l1 = d1 + bc[1];
        float mx = fmaxf(l0, l1);
        float lse = mx + logf(expf(l0 - mx) + expf(l1 - mx));
        out[(size_t)node * CLS + 0] = l0 - lse;
        out[(size_t)node * CLS + 1] = l1 - lse;
    }
}

extern "C" void kernel_launch(void* const* d_in, const int* in_sizes, int n_in,
                              void* d_out, int out_size, void* d_ws, size_t ws_size,
                              hipStream_t stream) {
    (void)in_sizes; (void)n_in; (void)out_size; (void)ws_size;
    const float* x     = (const float*)d_in[0];
    const int*   ei    = (const int*)  d_in[1];
    const float* W     = (const float*)d_in[2];
    const float* b     = (const float*)d_in[3];
    const float* gamma = (const float*)d_in[4];
    const float* beta  = (const float*)d_in[5];
    const float* rmean = (const float*)d_in[6];
    const float* rvar  = (const float*)d_in[7];
    const float* Wc    = (const float*)d_in[8];
    const float* bc    = (const float*)d_in[9];
    float* out = (float*)d_out;

    const int* src = ei;             // edge_index[0]
    const int* dst = ei + N_EDGES;   // edge_index[1]

    // workspace: dis[N] | bufA (hw, N*H) | bufB (h, N*H)  ~= 103 MB
    float* dis = (float*)d_ws;
    size_t off = (((size_t)N_NODES * 4) + 255) & ~(size_t)255;
    float* bufA = (float*)((char*)d_ws + off);
    float* bufB = bufA + (size_t)N_NODES * HDIM;

    const size_t NH = (size_t)N_NODES * HDIM;

    k_init_dis  <<<(N_NODES + 255) / 256, 256, 0, stream>>>(dis);
    k_deg       <<<(N_EDGES + 255) / 256, 256, 0, stream>>>(dst, dis);
    k_finish_dis<<<(N_NODES + 255) / 256, 256, 0, stream>>>(dis);

    const float* hin = x;
    for (int l = 0; l < LAYERS; ++l) {
        const float* Wl = W + (size_t)l * HDIM * HDIM;
        k_gemm128<MT_MAIN><<<FULL_BLOCKS, 256, 0, stream>>>(hin, Wl, bufA, 0);
        k_gemm128<REM_TILES><<<1, 256, 0, stream>>>(hin, Wl, bufA, FULL_BLOCKS * MT_MAIN);
        k_init_next<<<(int)((NH + 255) / 256), 256, 0, stream>>>(bufA, dis, b + l * HDIM, bufB);
        k_scatter  <<<(N_EDGES * 32) / 256, 256, 0, stream>>>(src, dst, dis, bufA, bufB);
        k_bn_relu  <<<(int)((NH + 255) / 256), 256, 0, stream>>>(bufB, rmean + l * HDIM,
                                                                 rvar + l * HDIM,
                                                                 gamma + l * HDIM,
                                                                 beta + l * HDIM);
        hin = bufB;
    }
    k_classify<<<(N_NODES * 32 + 255) / 256, 256, 0, stream>>>(bufB, Wc, bc, out);
}